// alt_power_whitening_68908455297218
// MI455X (gfx1250) — compile-verified
//
#include <hip/hip_runtime.h>
#include <hip/hip_bf16.h>

typedef __attribute__((ext_vector_type(2))) float v2f;
typedef __attribute__((ext_vector_type(4))) float v4f;
typedef __attribute__((ext_vector_type(8))) float v8f;
typedef __attribute__((ext_vector_type(4))) int   v4i;

#define GRAM_BLOCKS 256
#define PART_STRIDE 4160           // 4096 gram + 64 colsum per block partial
#define WS_GRAM   0
#define WS_COLSUM 4096
#define WS_MEAN   4160
#define WS_TVEC   4224
#define WS_PART   4288
#define XS        68               // LDS row stride in floats: 16B aligned, good banking

#define GLOBAL_AS __attribute__((address_space(1)))
#define LDS_AS    __attribute__((address_space(3)))

#if __has_builtin(__builtin_amdgcn_global_load_async_to_lds_b32) && \
    __has_builtin(__builtin_amdgcn_global_load_async_to_lds_b128)
#define HAVE_ASYNC 1
#else
#define HAVE_ASYNC 0
#endif

#if __has_builtin(__builtin_amdgcn_s_wait_asynccnt)
#define WAIT_ASYNC() __builtin_amdgcn_s_wait_asynccnt(0)
#else
#define WAIT_ASYNC() asm volatile("s_wait_asynccnt 0" ::: "memory")
#endif

static __device__ __forceinline__ v8f wmma_f32_4(v2f a, v2f b, v8f c) {
  // D(16x16,f32) = A(16x4,f32) x B(4x16,f32) + C
  return __builtin_amdgcn_wmma_f32_16x16x4_f32(false, a, false, b, (short)0, c,
                                               false, false);
}

#if HAVE_ASYNC
static __device__ __forceinline__ void async_copy_b32(const float* gsrc,
                                                      float* lds_dst) {
  __builtin_amdgcn_global_load_async_to_lds_b32(
      (GLOBAL_AS int*)gsrc, (LDS_AS int*)lds_dst, 0, 0);
}
static __device__ __forceinline__ void async_copy_b128(const float* gsrc,
                                                       float* lds_dst) {
  __builtin_amdgcn_global_load_async_to_lds_b128(
      (GLOBAL_AS v4i*)gsrc, (LDS_AS v4i*)lds_dst, 0, 0);
}
#endif

// ---------------------------------------------------------------------------
// Pass 1: per-block partial of Gram = x^T x (64x64) and column sums.
// LDS holds the 64x64 chunk transposed: XsT[col*XS + row].
// Full chunks: transpose done by 4-byte async copies (per-lane LDS address).
// Wave w computes Gram row-tile i=w against col-tiles j=0..3 with f32 WMMA.
// ---------------------------------------------------------------------------
__global__ __launch_bounds__(128) void gram_kernel(const float* __restrict__ x,
                                                   float* __restrict__ ws,
                                                   int n, int nchunks) {
  __shared__ float XsT[64 * XS];
  const int t    = threadIdx.x;
  const int wave = t >> 5;
  const int lane = t & 31;
  const int half = lane >> 4;
  const int lm   = lane & 15;

  v8f acc[4] = {};
  float csum = 0.0f;

  for (int chunk = blockIdx.x; chunk < nchunks; chunk += GRAM_BLOCKS) {
    const bool full = ((long)chunk * 64 + 64) <= (long)n;
#if HAVE_ASYNC
    if (full) {
      const float* src = x + (long)chunk * 4096;
#pragma unroll
      for (int q = 0; q < 32; ++q) {           // element e = q*128 + t
        const int e   = q * 128 + t;
        const int row = e >> 6;
        const int col = e & 63;
        async_copy_b32(src + e, &XsT[col * XS + row]);
      }
      WAIT_ASYNC();
    } else
#endif
    {
      const int  row  = t >> 1;
      const int  colb = (t & 1) * 32;
      const long grow = (long)chunk * 64 + row;
      const float* src = x + grow * 64 + colb;
#pragma unroll
      for (int q = 0; q < 8; ++q) {
        v4f val = {};
        if (grow < n) val = *(const v4f*)(src + 4 * q);
#pragma unroll
        for (int j = 0; j < 4; ++j)
          XsT[(colb + 4 * q + j) * XS + row] = val[j];
      }
    }
    __syncthreads();

    if (t < 64) {                      // column sums from transposed tile
      const float* col = &XsT[t * XS];
#pragma unroll 8
      for (int r2 = 0; r2 < 64; ++r2) csum += col[r2];
    }

#pragma unroll 1
    for (int k = 0; k < 64; k += 4) {  // K = sample dimension, 4 per WMMA
      const int ko = k + 2 * half;     // A/B frag: K = 2*half + v
      v2f a = *(const v2f*)&XsT[(16 * wave + lm) * XS + ko];
#pragma unroll
      for (int j = 0; j < 4; ++j) {
        v2f b = *(const v2f*)&XsT[(16 * j + lm) * XS + ko];
        acc[j] = wmma_f32_4(a, b, acc[j]);
      }
    }
    __syncthreads();
  }

  float* part = ws + WS_PART + (size_t)blockIdx.x * PART_STRIDE;
#pragma unroll
  for (int r = 0; r < 8; ++r) {        // C/D layout: M = r + 8*half, N = lm
    const int gr = 16 * wave + 8 * half + r;
#pragma unroll
    for (int j = 0; j < 4; ++j)
      part[gr * 64 + 16 * j + lm] = acc[j][r];
  }
  if (t < 64) part[4096 + t] = csum;
}

// ---------------------------------------------------------------------------
// Pass 2: deterministic fixed-order reduction of the block partials.
// ---------------------------------------------------------------------------
__global__ void reduce_kernel(float* __restrict__ ws) {
  const int i = blockIdx.x * blockDim.x + threadIdx.x;
  if (i >= PART_STRIDE) return;
  float s = 0.0f;
  const float* p = ws + WS_PART + i;
  for (int b = 0; b < GRAM_BLOCKS; ++b) s += p[(size_t)b * PART_STRIDE];
  ws[i] = s;   // gram at [0,4096), colsum at [4096,4160)
}

// ---------------------------------------------------------------------------
// Pass 3: serial eigensolve. One block of 64 threads; thread r owns row r of
// C and W in registers. v broadcast through LDS.
// ---------------------------------------------------------------------------
__global__ __launch_bounds__(64) void solve_kernel(const float* __restrict__ R,
                                                   float* __restrict__ ws,
                                                   float* __restrict__ out,
                                                   int n) {
  __shared__ float msh[64];
  __shared__ float vv[64];
  __shared__ float yv[64];
  const int r = threadIdx.x;
  const float invn = 1.0f / (float)n;

  float creg[64], wreg[64];

  const float m_r = ws[WS_COLSUM + r] * invn;
  msh[r] = m_r;
  ws[WS_MEAN + r] = m_r;
  __syncthreads();

#pragma unroll
  for (int c = 0; c < 64; ++c) {       // C0 = Gram/n - mean mean^T (exact)
    creg[c] = ws[WS_GRAM + r * 64 + c] * invn - m_r * msh[c];
    wreg[c] = 0.0f;
  }

#pragma unroll 1
  for (int col = 0; col < 64; ++col) {
    vv[r] = R[r * 64 + col];           // v0 = column `col` of R
    __syncthreads();
    float l = 0.0f;
#pragma unroll 1
    for (int it = 0; it <= 50; ++it) { // 50 normalized steps + eigenvalue pass
      float y = 0.0f;
#pragma unroll
      for (int k = 0; k < 64; ++k) y = fmaf(creg[k], vv[k], y);
      yv[r] = y;
      __syncthreads();
      float s = 0.0f;
#pragma unroll
      for (int k = 0; k < 64; ++k) { const float q = yv[k]; s = fmaf(q, q, s); }
      const float nrm = sqrtf(s);
      if (it < 50) vv[r] = y / nrm;
      else         l = nrm;
      __syncthreads();
    }
    const float vr = vv[r];
    const float ac = l * vr;
    const float aw = vr / sqrtf(l);
#pragma unroll
    for (int k = 0; k < 64; ++k) {
      const float vk = vv[k];
      creg[k] = fmaf(-ac, vk, creg[k]);   // C -= l v v^T
      wreg[k] = fmaf(aw, vk, wreg[k]);    // W += v v^T / sqrt(l)
    }
    __syncthreads();
  }

  float* outW = out + (size_t)n * 64;
  float* outM = outW + 4096;
  float* outC = outM + 64;
  float tv = 0.0f;
#pragma unroll
  for (int k = 0; k < 64; ++k) {
    outW[r * 64 + k] = wreg[k];
    outC[r * 64 + k] = creg[k];
    tv = fmaf(wreg[k], msh[k], tv);       // t = W @ mean
  }
  ws[WS_TVEC + r] = tv;
  outM[r] = 0.0f;                         // xc.mean == 0 (reference ~1e-8)
}

// ---------------------------------------------------------------------------
// Pass 4: whitened = xc @ W^T = x @ W^T - t,  t = W @ mean  (exact identity).
// Raw x feeds the GEMM; the mean correction is the per-column t subtraction.
// Wave w owns sample-row tile ib=w; A[m][K] = x[row][k+K], B[K][n] = W[16j+n][k+K].
// ---------------------------------------------------------------------------
__global__ __launch_bounds__(128) void whiten_kernel(const float* __restrict__ x,
                                                     const float* __restrict__ ws,
                                                     float* __restrict__ out,
                                                     int n) {
  __shared__ float Xs[64 * XS];
  __shared__ float Wsh[64 * XS];
  __shared__ float tsh[64];
  const int t    = threadIdx.x;
  const int wave = t >> 5;
  const int lane = t & 31;
  const int half = lane >> 4;
  const int lm   = lane & 15;
  const float* Wsrc = out + (size_t)n * 64;

  if (t < 64) tsh[t] = ws[WS_TVEC + t];

  const int chunk = blockIdx.x;
  const bool full = ((long)chunk * 64 + 64) <= (long)n;
#if HAVE_ASYNC
  if (full) {
    const float* src = x + (long)chunk * 4096;
#pragma unroll
    for (int q = 0; q < 8; ++q) {          // float4 element e4 = q*128 + t
      const int e4  = q * 128 + t;
      const int row = e4 >> 4;
      const int c4  = e4 & 15;
      async_copy_b128(src + e4 * 4, &Xs[row * XS + c4 * 4]);
      async_copy_b128(Wsrc + e4 * 4, &Wsh[row * XS + c4 * 4]);
    }
    WAIT_ASYNC();
  } else
#endif
  {
    const int  row  = t >> 1;
    const int  colb = (t & 1) * 32;
    const long grow = (long)chunk * 64 + row;
    const float* src = x + grow * 64 + colb;
#pragma unroll
    for (int q = 0; q < 8; ++q) {
      v4f val = {};
      if (grow < n) val = *(const v4f*)(src + 4 * q);
      *(v4f*)&Xs[row * XS + colb + 4 * q] = val;
      *(v4f*)&Wsh[row * XS + colb + 4 * q] =
          *(const v4f*)(Wsrc + row * 64 + colb + 4 * q);
    }
  }
  __syncthreads();

  v8f acc[4] = {};
#pragma unroll 1
  for (int k = 0; k < 64; k += 4) {      // K = feature dimension
    const int ko = k + 2 * half;
    v2f a = *(const v2f*)&Xs[(16 * wave + lm) * XS + ko];
#pragma unroll
    for (int j = 0; j < 4; ++j) {
      v2f b = *(const v2f*)&Wsh[(16 * j + lm) * XS + ko];
      acc[j] = wmma_f32_4(a, b, acc[j]);
    }
  }

#pragma unroll
  for (int r2 = 0; r2 < 8; ++r2) {
    const long row = (long)chunk * 64 + 16 * wave + 8 * half + r2;
    if (row < n) {
#pragma unroll
      for (int j = 0; j < 4; ++j)
        out[row * 64 + 16 * j + lm] = acc[j][r2] - tsh[16 * j + lm];
    }
  }
}

extern "C" void kernel_launch(void* const* d_in, const int* in_sizes, int n_in,
                              void* d_out, int out_size, void* d_ws, size_t ws_size,
                              hipStream_t stream) {
  const float* x = (const float*)d_in[0];
  const float* R = (const float*)d_in[1];
  float* out = (float*)d_out;
  float* ws  = (float*)d_ws;
  const int n = in_sizes[0] / 64;
  const int nchunks = (n + 63) / 64;

  gram_kernel<<<GRAM_BLOCKS, 128, 0, stream>>>(x, ws, n, nchunks);
  reduce_kernel<<<(PART_STRIDE + 255) / 256, 256, 0, stream>>>(ws);
  solve_kernel<<<1, 64, 0, stream>>>(R, ws, out, n);
  whiten_kernel<<<nchunks, 128, 0, stream>>>(x, ws, out, n);
}